// HeteroGNN_GAT_76613626626592
// MI455X (gfx1250) — compile-verified
//
#include <hip/hip_runtime.h>
#include <hip/hip_bf16.h>

// ---------------------------------------------------------------------------
// HeteroGNN (GAT) forward for MI455X / gfx1250.
// fp32 end-to-end; dense projections use V_WMMA_F32_16X16X4_F32.
// ---------------------------------------------------------------------------

#define N_NODES 100000     // N_IN == N_OUT == 100000
#define E_EDGES 1000000
#define HID 64
#define NUM_GRAPHS 256

typedef float v2f __attribute__((ext_vector_type(2)));
typedef float v8f __attribute__((ext_vector_type(8)));

// ---------------------------------------------------------------------------
// Small helpers
// ---------------------------------------------------------------------------
__device__ __forceinline__ void atomicMaxF(float* addr, float val) {
    // ordered-int trick; valid for any init <= all values (we init to -3e38)
    if (val >= 0.0f) atomicMax((int*)addr, __float_as_int(val));
    else             atomicMin((unsigned int*)addr, __float_as_uint(val));
}

__global__ void fill_kernel(float* __restrict__ p, float v, int n) {
    int i = blockIdx.x * 256 + threadIdx.x;
    if (i < n) p[i] = v;
}

// ---------------------------------------------------------------------------
// WMMA fp32 GEMM:  H[N,64] = X[N,K] @ W[K,64]   (K in {16,64}, N % 16 == 0)
// One wave -> 16 rows x 64 cols (4 N-tiles of 16x16), K consumed 4 at a time.
// A frag (16x4 f32): lanes 0-15 hold K=k0,k0+1 ; lanes 16-31 hold K=k0+2,k0+3
// B frag (4x16 f32): v0 = W[k0+2*half][col], v1 = W[k0+2*half+1][col]
// D (16x16 f32): vgpr r -> row r + 8*(lane>=16), col = lane&15
// ---------------------------------------------------------------------------
__device__ __forceinline__ v8f wmma4(v2f a, v2f b, v8f c) {
    return __builtin_amdgcn_wmma_f32_16x16x4_f32(
        false, a, false, b, (short)0, c, false, false);
}

__global__ __launch_bounds__(256) void gemm_wmma_kernel(
    const float* __restrict__ X, const float* __restrict__ W,
    float* __restrict__ H, int N, int K) {
    const int lane  = threadIdx.x & 31;
    const int wave  = threadIdx.x >> 5;
    const int tile  = blockIdx.x * 8 + wave;
    const int ntile = N >> 4;
    if (tile >= ntile) return;            // wave-uniform exit: EXEC all-1s for WMMA
    const int lhalf = lane >> 4;          // 0 | 1
    const int lrow  = lane & 15;
    const int row0  = tile << 4;

    v8f acc0 = {0,0,0,0,0,0,0,0};
    v8f acc1 = {0,0,0,0,0,0,0,0};
    v8f acc2 = {0,0,0,0,0,0,0,0};
    v8f acc3 = {0,0,0,0,0,0,0,0};

    const float* arow = X + (size_t)(row0 + lrow) * K + (lhalf << 1);
    for (int k0 = 0; k0 < K; k0 += 4) {
        v2f a; a.x = arow[k0]; a.y = arow[k0 + 1];
        const float* bp = W + (size_t)(k0 + (lhalf << 1)) * HID + lrow;
        v2f b0; b0.x = bp[0];  b0.y = bp[HID];
        v2f b1; b1.x = bp[16]; b1.y = bp[HID + 16];
        v2f b2; b2.x = bp[32]; b2.y = bp[HID + 32];
        v2f b3; b3.x = bp[48]; b3.y = bp[HID + 48];
        acc0 = wmma4(a, b0, acc0);
        acc1 = wmma4(a, b1, acc1);
        acc2 = wmma4(a, b2, acc2);
        acc3 = wmma4(a, b3, acc3);
    }

    float* out = H + (size_t)(row0 + (lhalf << 3)) * HID + lrow;
#pragma unroll
    for (int r = 0; r < 8; ++r) {
        out[(size_t)r * HID +  0] = acc0[r];
        out[(size_t)r * HID + 16] = acc1[r];
        out[(size_t)r * HID + 32] = acc2[r];
        out[(size_t)r * HID + 48] = acc3[r];
    }
}

// ---------------------------------------------------------------------------
// Precompute folded attention vectors:
//   vsrc[k] = sum_j w_src[k][j]*att_src[j]  (k < K)
//   vdst[k] = sum_j w_dst[k][j]*att_dst[j]  (k < K)
//   weatt[k]= sum_j w_edge[k][j]*att_edge[j] (k < 8)
// ---------------------------------------------------------------------------
__global__ void prep_vecs_kernel(
    const float* __restrict__ w_src, const float* __restrict__ w_dst,
    const float* __restrict__ w_edge,
    const float* __restrict__ att_src, const float* __restrict__ att_dst,
    const float* __restrict__ att_edge,
    float* __restrict__ vsrc, float* __restrict__ vdst, float* __restrict__ weatt,
    int K) {
    int t = threadIdx.x;
    if (t < K) {
        float s = 0.f;
        for (int j = 0; j < HID; ++j) s += w_src[t * HID + j] * att_src[j];
        vsrc[t] = s;
    } else if (t >= 64 && t < 64 + K) {
        int k = t - 64; float s = 0.f;
        for (int j = 0; j < HID; ++j) s += w_dst[k * HID + j] * att_dst[j];
        vdst[k] = s;
    } else if (t >= 128 && t < 136) {
        int k = t - 128; float s = 0.f;
        for (int j = 0; j < HID; ++j) s += w_edge[k * HID + j] * att_edge[j];
        weatt[k] = s;
    }
}

// a[i] = dot(x[i, 0..K-1], v)   — one wave per row
__global__ __launch_bounds__(256) void rowdot_kernel(
    const float* __restrict__ x, const float* __restrict__ v,
    float* __restrict__ a, int N, int K) {
    int lane = threadIdx.x & 31;
    int row  = blockIdx.x * 8 + (threadIdx.x >> 5);
    if (row >= N) return;
    float p = 0.f;
    for (int j = lane; j < K; j += 32) p += x[(size_t)row * K + j] * v[j];
#pragma unroll
    for (int off = 16; off > 0; off >>= 1) p += __shfl_xor(p, off, 32);
    if (lane == 0) a[row] = p;
}

// ---------------------------------------------------------------------------
// Edge passes (segment softmax + scatter)
// ---------------------------------------------------------------------------
__global__ void edge_alpha_kernel(
    const int* __restrict__ src, const int* __restrict__ dst,
    const float* __restrict__ ea, const float* __restrict__ asrc,
    const float* __restrict__ adst, const float* __restrict__ weatt,
    float* __restrict__ alpha, float* __restrict__ m, int E) {
    int e = blockIdx.x * 256 + threadIdx.x;
    if (e >= E) return;
    int s = src[e], d = dst[e];
    float al = asrc[s] + adst[d];
    const float* p = ea + (size_t)e * 8;
#pragma unroll
    for (int k = 0; k < 8; ++k) al += p[k] * weatt[k];
    al = al > 0.f ? al : 0.2f * al;          // LeakyReLU(0.2)
    alpha[e] = al;
    atomicMaxF(m + d, al);
}

__global__ void edge_exp_kernel(
    const int* __restrict__ dst, float* __restrict__ alpha,
    const float* __restrict__ m, float* __restrict__ sseg, int E) {
    int e = blockIdx.x * 256 + threadIdx.x;
    if (e >= E) return;
    int d = dst[e];
    float w = expf(alpha[e] - m[d]);
    alpha[e] = w;
    atomicAdd(sseg + d, w);
}

// one wave per edge: acc[dst] += (alpha/s[dst]) * h_src[src]
__global__ __launch_bounds__(256) void edge_scatter_kernel(
    const int* __restrict__ src, const int* __restrict__ dst,
    const float* __restrict__ alpha, const float* __restrict__ sseg,
    const float* __restrict__ h, float* __restrict__ acc, int E) {
    int lane = threadIdx.x & 31;
    int e = blockIdx.x * 8 + (threadIdx.x >> 5);
    if (e >= E) return;
    int s = src[e], d = dst[e];
    float w = alpha[e] / sseg[d];
    const float* hp = h + (size_t)s * HID;
    float* ap = acc + (size_t)d * HID;
    atomicAdd(ap + lane,      w * hp[lane]);
    atomicAdd(ap + lane + 32, w * hp[lane + 32]);
}

// ---------------------------------------------------------------------------
// Combine edge types (mean + bias), accumulate BN stats
// ---------------------------------------------------------------------------
__global__ __launch_bounds__(256) void combine_stats_kernel(
    const float* __restrict__ acc, const float* __restrict__ biasA,
    const float* __restrict__ biasB, float invT, float* __restrict__ xout,
    float* __restrict__ gsum, float* __restrict__ gsq, int N) {
    int c  = threadIdx.x & 63;
    int rg = threadIdx.x >> 6;       // 0..3
    int row0 = blockIdx.x * 64;
    float b = biasA[c] + (biasB ? biasB[c] : 0.f);
    float s = 0.f, q = 0.f;
    for (int r = rg; r < 64; r += 4) {
        int i = row0 + r;
        if (i < N) {
            float v = (acc[(size_t)i * HID + c] + b) * invT;
            xout[(size_t)i * HID + c] = v;
            s += v; q += v * v;
        }
    }
    __shared__ float ls[256], lq[256];
    ls[threadIdx.x] = s; lq[threadIdx.x] = q;
    __syncthreads();
    if (rg == 0) {
        s = ls[c] + ls[c + 64] + ls[c + 128] + ls[c + 192];
        q = lq[c] + lq[c + 64] + lq[c + 128] + lq[c + 192];
        atomicAdd(gsum + c, s);
        atomicAdd(gsq + c, q);
    }
}

__global__ void bn_finalize_kernel(
    const float* __restrict__ gsum, const float* __restrict__ gsq,
    const float* __restrict__ gamma, const float* __restrict__ beta,
    float* __restrict__ scale, float* __restrict__ shift, float invN) {
    int c = threadIdx.x;
    if (c >= HID) return;
    float mu  = gsum[c] * invN;
    float var = gsq[c] * invN - mu * mu;
    float sc  = gamma[c] * rsqrtf(var + 1e-5f);
    scale[c] = sc;
    shift[c] = beta[c] - mu * sc;
}

__global__ void bn_apply_act_kernel(
    float* __restrict__ x, const float* __restrict__ scale,
    const float* __restrict__ shift, int total) {
    int i = blockIdx.x * 256 + threadIdx.x;
    if (i >= total) return;
    int c = i & 63;
    float v = x[i] * scale[c] + shift[c];
    x[i] = v > 0.f ? v : 0.01f * v;          // LeakyReLU(0.01)
}

// ---------------------------------------------------------------------------
// Graph pooling + MLP head
// ---------------------------------------------------------------------------
__global__ void pool_kernel(const float* __restrict__ x,
                            const int* __restrict__ batch,
                            float* __restrict__ pool, int N) {
    int i = blockIdx.x * 256 + threadIdx.x;
    if (i >= N * HID) return;
    int row = i >> 6, c = i & 63;
    atomicAdd(pool + (size_t)batch[row] * HID + c, x[i]);
}

__global__ __launch_bounds__(64) void mlp_kernel(
    const float* __restrict__ pool,
    const float* __restrict__ W0, const float* __restrict__ b0,
    const float* __restrict__ W1, const float* __restrict__ b1,
    const float* __restrict__ W2, const float* __restrict__ b2,
    float* __restrict__ out) {
    __shared__ float xr[64], h1[50], h2[25];
    int g = blockIdx.x, t = threadIdx.x;
    xr[t] = pool[(size_t)g * HID + t];
    __syncthreads();
    if (t < 50) {
        float s = b0[t];
        for (int k = 0; k < 64; ++k) s += xr[k] * W0[k * 50 + t];
        h1[t] = s > 0.f ? s : 0.f;
    }
    __syncthreads();
    if (t < 25) {
        float s = b1[t];
        for (int k = 0; k < 50; ++k) s += h1[k] * W1[k * 25 + t];
        h2[t] = s > 0.f ? s : 0.f;
    }
    __syncthreads();
    if (t == 0) {
        float s = b2[0];
        for (int k = 0; k < 25; ++k) s += h2[k] * W2[k];
        out[g] = s;
    }
}

// ---------------------------------------------------------------------------
// Host orchestration
// ---------------------------------------------------------------------------
static inline void fill_async(float* p, float v, int n, hipStream_t s) {
    fill_kernel<<<(n + 255) / 256, 256, 0, s>>>(p, v, n);
}

extern "C" void kernel_launch(void* const* d_in, const int* in_sizes, int n_in,
                              void* d_out, int out_size, void* d_ws, size_t ws_size,
                              hipStream_t stream) {
    const int N = N_NODES, E = E_EDGES;

    const float* x_inpin  = (const float*)d_in[0];
    const float* x_outpin = (const float*)d_in[1];
    const int*   ei[3]    = {(const int*)d_in[2], (const int*)d_in[4], (const int*)d_in[6]};
    const float* ea[3]    = {(const float*)d_in[3], (const float*)d_in[5], (const float*)d_in[7]};
    const int*   batch_in  = (const int*)d_in[8];
    const int*   batch_out = (const int*)d_in[9];

    // params pytree flatten order (dict keys sorted):
    //   layers[l]: bn_in(g,b), bn_out(g,b), then e1,e2,e3 each as
    //              att_dst, att_edge, att_src, bias, w_dst, w_edge, w_src
    //   mlp: (W0,b0),(W1,b1),(W2,b2)
    // layer base = 10 + 25*l ; edge-type group base = Lb + 4 + 7*t ; mlp base = 60

    // ---- workspace carve-up (floats) ----
    float* ws = (float*)d_ws;
    size_t o = 0;
    float* hsrc    = ws + o; o += (size_t)N * HID;
    float* a_src   = ws + o; o += N;
    float* a_dst   = ws + o; o += N;
    float* alpha   = ws + o; o += E;
    float* mseg    = ws + o; o += N;
    float* sseg    = ws + o; o += N;
    float* acc_in  = ws + o; o += (size_t)N * HID;
    float* acc_out = ws + o; o += (size_t)N * HID;
    float* xin_cur = ws + o; o += (size_t)N * HID;
    float* xout_cur= ws + o; o += (size_t)N * HID;
    float* gsum    = ws + o; o += HID;
    float* gsq     = ws + o; o += HID;
    float* scale   = ws + o; o += HID;
    float* shift   = ws + o; o += HID;
    float* vsrc    = ws + o; o += HID;
    float* vdst    = ws + o; o += HID;
    float* weatt   = ws + o; o += 8;
    float* pool    = ws + o; o += (size_t)NUM_GRAPHS * HID;
    (void)ws_size; (void)in_sizes; (void)n_in; (void)out_size;

    const int gemm_blocks    = ((N >> 4) + 7) / 8;   // 6250 tiles / 8 waves
    const int rowdot_blocks  = (N + 7) / 8;
    const int edge_blocks    = (E + 255) / 256;
    const int scatter_blocks = (E + 7) / 8;
    const int elem_blocks    = (N * HID + 255) / 256;
    const int comb_blocks    = (N + 63) / 64;

    for (int layer = 0; layer < 2; ++layer) {
        const int Lb = 10 + layer * 25;
        const int K  = (layer == 0) ? 16 : HID;
        const float* xi = (layer == 0) ? x_inpin  : xin_cur;
        const float* xo = (layer == 0) ? x_outpin : xout_cur;

        // edge type wiring: e1: outpin->inpin, e2: inpin->outpin, e3: inpin->inpin
        const float* srcx[3] = {xo, xi, xi};
        const float* dstx[3] = {xi, xo, xi};
        float*       accs[3] = {acc_in, acc_out, acc_in};

        fill_async(acc_in,  0.f, N * HID, stream);
        fill_async(acc_out, 0.f, N * HID, stream);

        for (int t = 0; t < 3; ++t) {
            const int g = Lb + 4 + 7 * t;
            const float* att_dst  = (const float*)d_in[g + 0];
            const float* att_edge = (const float*)d_in[g + 1];
            const float* att_src  = (const float*)d_in[g + 2];
            const float* w_dst    = (const float*)d_in[g + 4];
            const float* w_edge   = (const float*)d_in[g + 5];
            const float* w_src    = (const float*)d_in[g + 6];

            prep_vecs_kernel<<<1, 256, 0, stream>>>(
                w_src, w_dst, w_edge, att_src, att_dst, att_edge,
                vsrc, vdst, weatt, K);

            gemm_wmma_kernel<<<gemm_blocks, 256, 0, stream>>>(srcx[t], w_src, hsrc, N, K);

            rowdot_kernel<<<rowdot_blocks, 256, 0, stream>>>(srcx[t], vsrc, a_src, N, K);
            rowdot_kernel<<<rowdot_blocks, 256, 0, stream>>>(dstx[t], vdst, a_dst, N, K);

            fill_async(mseg, -3.0e38f, N, stream);
            fill_async(sseg, 0.f, N, stream);

            const int* src = ei[t];
            const int* dst = ei[t] + E;
            edge_alpha_kernel<<<edge_blocks, 256, 0, stream>>>(
                src, dst, ea[t], a_src, a_dst, weatt, alpha, mseg, E);
            edge_exp_kernel<<<edge_blocks, 256, 0, stream>>>(dst, alpha, mseg, sseg, E);
            edge_scatter_kernel<<<scatter_blocks, 256, 0, stream>>>(
                src, dst, alpha, sseg, hsrc, accs[t], E);
        }

        const float* bias_e1 = (const float*)d_in[Lb + 4 + 0 * 7 + 3];
        const float* bias_e2 = (const float*)d_in[Lb + 4 + 1 * 7 + 3];
        const float* bias_e3 = (const float*)d_in[Lb + 4 + 2 * 7 + 3];

        // inpin: mean over {e1, e3}
        fill_async(gsum, 0.f, HID, stream);
        fill_async(gsq,  0.f, HID, stream);
        combine_stats_kernel<<<comb_blocks, 256, 0, stream>>>(
            acc_in, bias_e1, bias_e3, 0.5f, xin_cur, gsum, gsq, N);
        bn_finalize_kernel<<<1, 64, 0, stream>>>(
            gsum, gsq, (const float*)d_in[Lb + 0], (const float*)d_in[Lb + 1],
            scale, shift, 1.0f / N);
        bn_apply_act_kernel<<<elem_blocks, 256, 0, stream>>>(xin_cur, scale, shift, N * HID);

        // outpin: only e2
        fill_async(gsum, 0.f, HID, stream);
        fill_async(gsq,  0.f, HID, stream);
        combine_stats_kernel<<<comb_blocks, 256, 0, stream>>>(
            acc_out, bias_e2, nullptr, 1.0f, xout_cur, gsum, gsq, N);
        bn_finalize_kernel<<<1, 64, 0, stream>>>(
            gsum, gsq, (const float*)d_in[Lb + 2], (const float*)d_in[Lb + 3],
            scale, shift, 1.0f / N);
        bn_apply_act_kernel<<<elem_blocks, 256, 0, stream>>>(xout_cur, scale, shift, N * HID);
    }

    // pooling
    fill_async(pool, 0.f, NUM_GRAPHS * HID, stream);
    pool_kernel<<<elem_blocks, 256, 0, stream>>>(xin_cur,  batch_in,  pool, N);
    pool_kernel<<<elem_blocks, 256, 0, stream>>>(xout_cur, batch_out, pool, N);

    // MLP head -> d_out [256]
    mlp_kernel<<<NUM_GRAPHS, 64, 0, stream>>>(
        pool,
        (const float*)d_in[60], (const float*)d_in[61],
        (const float*)d_in[62], (const float*)d_in[63],
        (const float*)d_in[64], (const float*)d_in[65],
        (float*)d_out);
}